// Bati_89799176225034
// MI455X (gfx1250) — compile-verified
//
#include <hip/hip_runtime.h>
#include <hip/hip_bf16.h>

// ---------------------------------------------------------------------------
// Problem constants
// ---------------------------------------------------------------------------
#define NB   256    // batch
#define TT   2000   // timesteps
#define HID  51     // hidden of LSTM1
#define GSZ  204    // 4*HID gate vector
#define FEAT 49     // 7x7 input patch

// WMMA tiling: M-tile = 16 samples per workgroup, N = 208 (13 tiles of 16),
// K = 52 (51 padded). 13 waves per workgroup, one N-tile per wave.
#define NTILES   13
#define THREADS  (NTILES * 32)   // 416
#define KSTEPS   13              // 52 / 4
#define H1STRIDE 52              // h1 LDS row stride (floats), conflict-free
#define GSTRIDE  212             // gates LDS row stride (floats), conflict-free

typedef __attribute__((ext_vector_type(2))) float v2f;
typedef __attribute__((ext_vector_type(8))) float v8f;

__device__ __forceinline__ float sigf(float x) {
    return 1.0f / (1.0f + __expf(-x));
}
__device__ __forceinline__ float tanhfast(float x) {
    return 2.0f / (1.0f + __expf(-2.0f * x)) - 1.0f;
}

// ---------------------------------------------------------------------------
// Pass 1: x_lin[t][n] = relu(stimulus[n][t][:] . Wl + bl)   (parallel over N*T)
// ---------------------------------------------------------------------------
__global__ void xlin_kernel(const float* __restrict__ stim,
                            const float* __restrict__ Wl,
                            const float* __restrict__ bl,
                            float* __restrict__ xl) {
    int idx = blockIdx.x * blockDim.x + threadIdx.x;   // idx = n*TT + t
    if (idx >= NB * TT) return;
    int n = idx / TT;
    int t = idx - n * TT;
    const float* s = stim + (size_t)idx * FEAT;
    float acc = bl[0];
#pragma unroll
    for (int f = 0; f < FEAT; ++f) acc = fmaf(s[f], Wl[f], acc);
    xl[t * NB + n] = fmaxf(acc, 0.0f);
}

// ---------------------------------------------------------------------------
// Pass 2: recurrent loop. One workgroup = 16 samples, 13 waves.
// Per step: gates(16x204) = h1(16x51) @ Whh1^T(51x204) via f32 WMMA 16x16x4,
// then LSTM1 elementwise, LSTM2 (51->1), output affine.
// ---------------------------------------------------------------------------
__global__ void __launch_bounds__(THREADS)
lstm_seq_kernel(const float* __restrict__ xl,     // [TT][NB]
                const float* __restrict__ Wih1,   // [204]
                const float* __restrict__ Whh1,   // [204][51]
                const float* __restrict__ bih1,   // [204]
                const float* __restrict__ bhh1,   // [204]
                const float* __restrict__ Wih2,   // [4][51]
                const float* __restrict__ Whh2,   // [4]
                const float* __restrict__ bih2,   // [4]
                const float* __restrict__ bhh2,   // [4]
                const float* __restrict__ Wa,     // [1]
                const float* __restrict__ ba,     // [1]
                float* __restrict__ out)          // [NB][TT]
{
    __shared__ __align__(16) float h1s[16 * H1STRIDE];     // h1 state, k padded to 52 (col 51 = 0)
    __shared__ __align__(16) float gates_s[16 * GSTRIDE];  // 16 x 208 gate matrix
    __shared__ float wih2s[GSZ];                           // Wih2 rows
    __shared__ float g2s[64];                              // LSTM2 pre-activations
    __shared__ float xl_sh[16];
    __shared__ float h2_sh[16];
    __shared__ float whh2s[4];
    __shared__ float b2s[4];

    const int tid  = threadIdx.x;
    const int lane = tid & 31;
    const int wv   = tid >> 5;          // wave id == N-tile id, 0..12
    const int n0   = blockIdx.x * 16;   // first sample of this block

    // ---- one-time init -----------------------------------------------------
    for (int i = tid; i < 16 * H1STRIDE; i += THREADS) h1s[i] = 0.0f;
    for (int i = tid; i < GSZ; i += THREADS)           wih2s[i] = Wih2[i];
    if (tid < 4) {
        whh2s[tid] = Whh2[tid];
        b2s[tid]   = bih2[tid] + bhh2[tid];
    }
    if (tid < 16) h2_sh[tid] = 0.0f;

    // B operand (Whh1^T column block) held in registers for the whole run.
    // B layout for V_WMMA_F32_16X16X4_F32 (4x16, KxN):
    //   lane l, vgpr v  ->  element (K = 2*(l>>4) + v, N = l&15)
    const int colB = wv * 16 + (lane & 15);
    const int krow = 2 * (lane >> 4);
    v2f Breg[KSTEPS];
#pragma unroll
    for (int ks = 0; ks < KSTEPS; ++ks) {
        int k = 4 * ks + krow;
        float bx = (colB < GSZ && k     < HID) ? Whh1[colB * HID + k    ] : 0.0f;
        float by = (colB < GSZ && k + 1 < HID) ? Whh1[colB * HID + k + 1] : 0.0f;
        Breg[ks].x = bx;
        Breg[ks].y = by;
    }

    // Per-thread LSTM1 cell ownership: units u = n*51 + j, 816 total.
    // Thread owns uA = tid and (if valid) uB = tid + 416. c1 lives in registers.
    const int uA = tid;
    const int uB = tid + THREADS;
    const int nA = uA / HID, jA = uA - nA * HID;
    const int nB = (uB < 16 * HID) ? (uB / HID) : 0;
    const int jB = (uB < 16 * HID) ? (uB - nB * HID) : 0;
    const bool hasB = (uB < 16 * HID);
    float wihA[4], btA[4], wihB[4], btB[4];
#pragma unroll
    for (int s2 = 0; s2 < 4; ++s2) {
        int gA = jA + HID * s2;
        wihA[s2] = Wih1[gA];
        btA[s2]  = bih1[gA] + bhh1[gA];
        int gB = jB + HID * s2;
        wihB[s2] = hasB ? Wih1[gB] : 0.0f;
        btB[s2]  = hasB ? (bih1[gB] + bhh1[gB]) : 0.0f;
    }
    float c1a = 0.0f, c1b = 0.0f;   // LSTM1 cell state
    float c2  = 0.0f;               // LSTM2 cell state (threads 0..15)
    const float waR = Wa[0];
    const float baR = ba[0];

    __syncthreads();

    const int rowA = lane & 15;
    const int row0 = 8 * (lane >> 4);
    const int colD = wv * 16 + (lane & 15);

    // ---- sequential time loop ---------------------------------------------
    for (int t = 0; t < TT; ++t) {
        // prefetch this step's input projection (overlaps with WMMA below)
        float xlv = 0.0f;
        if (tid < 16) xlv = xl[t * NB + n0 + tid];

        // ---- gates = h1 @ Whh1^T via 13 chained f32 WMMAs per wave ----
        v8f acc = (v8f){0.f, 0.f, 0.f, 0.f, 0.f, 0.f, 0.f, 0.f};
#pragma unroll
        for (int ks = 0; ks < KSTEPS; ++ks) {
            // A layout (16x4, MxK): lane l, vgpr v -> (M = l&15, K = 2*(l>>4)+v)
            v2f a = *(const v2f*)&h1s[rowA * H1STRIDE + krow + 4 * ks];
            acc = __builtin_amdgcn_wmma_f32_16x16x4_f32(
                false, a, false, Breg[ks], (short)0, acc, false, false);
        }
#pragma unroll
        for (int v = 0; v < 8; ++v)
            gates_s[(row0 + v) * GSTRIDE + colD] = acc[v];

        if (tid < 16) xl_sh[tid] = xlv;
        __syncthreads();

        // ---- LSTM1 elementwise update; write new h1 into LDS ----
        {
            float xn = xl_sh[nA];
            float gi = gates_s[nA * GSTRIDE + jA          ] + xn * wihA[0] + btA[0];
            float gf = gates_s[nA * GSTRIDE + jA + HID    ] + xn * wihA[1] + btA[1];
            float gg = gates_s[nA * GSTRIDE + jA + 2 * HID] + xn * wihA[2] + btA[2];
            float go = gates_s[nA * GSTRIDE + jA + 3 * HID] + xn * wihA[3] + btA[3];
            float iv = sigf(gi), fv = sigf(gf), gv = tanhfast(gg), ov = sigf(go);
            c1a = fv * c1a + iv * gv;
            h1s[nA * H1STRIDE + jA] = ov * tanhfast(c1a);
        }
        if (hasB) {
            float xn = xl_sh[nB];
            float gi = gates_s[nB * GSTRIDE + jB          ] + xn * wihB[0] + btB[0];
            float gf = gates_s[nB * GSTRIDE + jB + HID    ] + xn * wihB[1] + btB[1];
            float gg = gates_s[nB * GSTRIDE + jB + 2 * HID] + xn * wihB[2] + btB[2];
            float go = gates_s[nB * GSTRIDE + jB + 3 * HID] + xn * wihB[3] + btB[3];
            float iv = sigf(gi), fv = sigf(gf), gv = tanhfast(gg), ov = sigf(go);
            c1b = fv * c1b + iv * gv;
            h1s[nB * H1STRIDE + jB] = ov * tanhfast(c1b);
        }
        __syncthreads();

        // ---- LSTM2 pre-activations: 64 threads, one (sample, gate) each ----
        if (tid < 64) {
            int n = tid >> 2, q = tid & 3;
            float s = b2s[q] + h2_sh[n] * whh2s[q];
            const float* hr = &h1s[n * H1STRIDE];
            const float* wr = &wih2s[q * HID];
#pragma unroll
            for (int j = 0; j < HID; ++j) s = fmaf(hr[j], wr[j], s);
            g2s[tid] = s;
        }
        __syncthreads();

        // ---- LSTM2 cell update + output affine: 16 threads ----
        if (tid < 16) {
            float iv = sigf(g2s[tid * 4 + 0]);
            float fv = sigf(g2s[tid * 4 + 1]);
            float gv = tanhfast(g2s[tid * 4 + 2]);
            float ov = sigf(g2s[tid * 4 + 3]);
            c2 = fv * c2 + iv * gv;
            float h2 = ov * tanhfast(c2);
            h2_sh[tid] = h2;
            out[(size_t)(n0 + tid) * TT + t] = fmaf(h2, waR, baR);
        }
        __syncthreads();
    }
}

// ---------------------------------------------------------------------------
// Launch
// ---------------------------------------------------------------------------
extern "C" void kernel_launch(void* const* d_in, const int* in_sizes, int n_in,
                              void* d_out, int out_size, void* d_ws, size_t ws_size,
                              hipStream_t stream) {
    const float* stim = (const float*)d_in[0];
    const float* Wl   = (const float*)d_in[1];
    const float* bl   = (const float*)d_in[2];
    const float* Wih1 = (const float*)d_in[3];
    const float* Whh1 = (const float*)d_in[4];
    const float* bih1 = (const float*)d_in[5];
    const float* bhh1 = (const float*)d_in[6];
    const float* Wih2 = (const float*)d_in[7];
    const float* Whh2 = (const float*)d_in[8];
    const float* bih2 = (const float*)d_in[9];
    const float* bhh2 = (const float*)d_in[10];
    const float* Wa   = (const float*)d_in[11];
    const float* ba   = (const float*)d_in[12];
    float* out = (float*)d_out;

    // Workspace: x_lin laid out [T][N] = 512000 floats (2 MB)
    float* xl = (float*)d_ws;

    {
        int total = NB * TT;
        int blk = 256;
        xlin_kernel<<<(total + blk - 1) / blk, blk, 0, stream>>>(stim, Wl, bl, xl);
    }
    lstm_seq_kernel<<<NB / 16, THREADS, 0, stream>>>(
        xl, Wih1, Whh1, bih1, bhh1, Wih2, Whh2, bih2, bhh2, Wa, ba, out);
}